// SSIM_15418932592755
// MI455X (gfx1250) — compile-verified
//
#include <hip/hip_runtime.h>
#include <cstdint>

// ---------------------------------------------------------------------------
// Fused SSIM for MI455X (gfx1250, wave32).
//
// Five separable 11-tap Gaussian blurs (X, Y, X*X, Y*Y, X*Y) with VALID
// padding on 16x3x768x768 f32, then elementwise SSIM map + per-batch mean.
// Memory floor: 226 MB input -> ~10us @ 23.3 TB/s, so everything is fused in
// one pass through the inputs; tile halos are re-read from the 192 MB L2.
// The 11-tap conv is expressed as banded 16x16xK f32 matmuls on
// V_WMMA_F32_16X16X4_F32 (7 K-chunks per pass), and the input tiles are
// fetched by the Tensor Data Mover (tensor_load_to_lds, 6-arg toolchain form).
// ---------------------------------------------------------------------------

typedef __attribute__((ext_vector_type(2))) float        v2f;
typedef __attribute__((ext_vector_type(8))) float        v8f;
typedef __attribute__((ext_vector_type(4))) unsigned int v4u;
typedef __attribute__((ext_vector_type(8))) int          v8i;
typedef __attribute__((ext_vector_type(4))) int          v4i;

namespace {
constexpr int WIN   = 11;
constexpr int HALO  = WIN - 1;        // 10
constexpr int HIMG  = 768, WIMG = 768;
constexpr int HO    = HIMG - HALO;    // 758
constexpr int WO    = WIMG - HALO;    // 758
constexpr int BATCH = 16;
constexpr int CHAN  = 3;

constexpr int TM    = 16;             // output rows per workgroup
constexpr int TN    = 16;             // output cols per wave
constexpr int NWAVE = 8;              // 256 threads = 8 wave32
constexpr int WGN   = NWAVE * TN;     // 128 output cols per workgroup
constexpr int IR    = TM + HALO;      // 26 input rows
constexpr int IC    = WGN + HALO;     // 138 input cols (TDM row stride in LDS)
constexpr int KCH   = 7;              // ceil(26/4) K-chunks (K padded to 28)

constexpr int SH    = 140;            // staging stride (2*SH%64=24, 8*SH%64=32)
constexpr int HR    = 26;             // staging rows (pad K rows are clamped, not stored)

constexpr int TILES_X = (WO + WGN - 1) / WGN;  // 6
constexpr int TILES_Y = (HO + TM  - 1) / TM;   // 48
constexpr int NIMG    = BATCH * CHAN;          // 48
constexpr int TOTROWS = NIMG * HIMG;           // 36864 rows in the whole stack
}

// Banded Gaussian weight: w[d] for d in [0,11), else 0 (clamped safe read).
__device__ __forceinline__ float bandw(const float* __restrict__ w, int d) {
    int cl = d < 0 ? 0 : (d > WIN - 1 ? WIN - 1 : d);
    float v = w[cl];
    return (d >= 0 && d < WIN) ? v : 0.0f;
}

#if __has_builtin(__builtin_amdgcn_tensor_load_to_lds)
#define SSIM_USE_TDM 1
// Issue one 2-D TDM transfer: IC x IR f32 tile, rows stored contiguously in
// LDS (stride IC).  tensor_dim0/1 are the valid extents measured from the
// tile origin, so edge-tile overreads resolve as OOB -> zero fill (finite,
// and only ever feeding masked outputs).
__device__ __forceinline__ void tdm_load_tile(const float* gsrc, unsigned lds_addr,
                                              int rem_cols, int rem_rows) {
    uint64_t ga = (uint64_t)(uintptr_t)gsrc;
    v4u g0;
    g0[0] = 1u;                                   // count=1 (valid user D#)
    g0[1] = lds_addr;                             // LDS byte address
    g0[2] = (unsigned)(ga & 0xFFFFFFFFu);         // global_addr[31:0]
    g0[3] = (unsigned)((ga >> 32) & 0x01FFFFFFu)  // global_addr[56:32]
          | (2u << 30);                           // type=2 ("image")
    v8i g1;
    g1[0] = (int)(2u << 16);                      // data_size=4B; mask/flags=0
    g1[1] = (int)(((unsigned)rem_cols & 0xFFFFu) << 16);            // dim0[15:0]
    g1[2] = (int)((((unsigned)rem_cols >> 16) & 0xFFFFu)
          | (((unsigned)rem_rows & 0xFFFFu) << 16));                // dim0 hi / dim1 lo
    g1[3] = (int)((((unsigned)rem_rows >> 16) & 0xFFFFu)
          | ((unsigned)IC << 16));                                  // dim1 hi / tile_dim0=138
    g1[4] = (int)(IR & 0xFFFF);                   // tile_dim1=26, tile_dim2=0
    g1[5] = WIMG;                                 // tensor_dim0_stride = 768 (lo32)
    g1[6] = 0;                                    // stride0 hi16, dim1_stride lo16
    g1[7] = 0;
    v4i z4 = {0, 0, 0, 0};                        // groups 2/3 unused (2-D tensor)
    v8i z8 = {0, 0, 0, 0, 0, 0, 0, 0};            // extra group (6-arg toolchain form)
    __builtin_amdgcn_tensor_load_to_lds(g0, g1, z4, z4, z8, 0);
}
#else
#define SSIM_USE_TDM 0
#endif

__global__ __launch_bounds__(256)
void ssim_tile_kernel(const float* __restrict__ X, const float* __restrict__ Y,
                      const float* __restrict__ win, float* __restrict__ partial) {
    __shared__ float lX[IR * IC];
    __shared__ float lY[IR * IC];
    __shared__ float lH[HR * SH];       // per-wave private 16-col staging strips
    __shared__ float wavesum[NWAVE];

    const int img  = blockIdx.z;                  // b*3 + c
    const int row0 = blockIdx.y * TM;
    const int col0 = blockIdx.x * WGN;
    const int absrow = img * HIMG + row0;         // row within the whole stack

#if SSIM_USE_TDM
    if (threadIdx.x < 32) {                       // one DMA issue per workgroup
        const size_t goff = (size_t)absrow * WIMG + col0;
        tdm_load_tile(X + goff, (unsigned)(uintptr_t)&lX[0],
                      WIMG - col0, TOTROWS - absrow);
        tdm_load_tile(Y + goff, (unsigned)(uintptr_t)&lY[0],
                      WIMG - col0, TOTROWS - absrow);
        __builtin_amdgcn_s_wait_tensorcnt((short)0);
    }
    __syncthreads();
#else
    {   // cooperative clamped tile load (fallback path)
        const size_t base = (size_t)img * HIMG * WIMG;
        for (int i = threadIdx.x; i < IR * IC; i += 256) {
            int r = i / IC, c = i - r * IC;
            int gr = row0 + r; if (gr > HIMG - 1) gr = HIMG - 1;
            int gc = col0 + c; if (gc > WIMG - 1) gc = WIMG - 1;
            size_t g = base + (size_t)gr * WIMG + gc;
            lX[i] = X[g];
            lY[i] = Y[g];
        }
        __syncthreads();
    }
#endif

    const int lane = threadIdx.x & 31;
    const int wid  = threadIdx.x >> 5;
    const int half = lane >> 4;          // which 16-lane half (K pairs 0,1 vs 2,3)
    const int lm   = lane & 15;
    const int cb   = wid * TN;           // this wave's column base in the tile

    // Banded weight fragments: value = w[k - lm].  The same fragment is the
    // B-matrix of the horizontal pass (lm = N) and the A-matrix of the
    // vertical pass (lm = M).  K positions 26/27 get weight 0.
    v2f wb[KCH];
#pragma unroll
    for (int kc = 0; kc < KCH; ++kc) {
        int k = kc * 4 + 2 * half;
        wb[kc].x = bandw(win, k - lm);
        wb[kc].y = bandw(win, k + 1 - lm);
    }

    // Cache X/Y A-fragments once: two overlapping 16-row M-blocks (rows 0..15
    // and 10..25).  Zero-weight K-pad positions (k>25) are clamped to k=25 so
    // every LDS read hits initialized, finite data (NaN * 0 would poison WMMA).
    v2f xa[2][KCH], ya[2][KCH];
#pragma unroll
    for (int blk = 0; blk < 2; ++blk) {
        const int rb = blk * 10;
#pragma unroll
        for (int kc = 0; kc < KCH; ++kc) {
            int k  = kc * 4 + 2 * half;
            int k0 = k     > 25 ? 25 : k;
            int k1 = k + 1 > 25 ? 25 : k + 1;
            int o  = (rb + lm) * IC + cb;
            xa[blk][kc].x = lX[o + k0];
            xa[blk][kc].y = lX[o + k1];
            ya[blk][kc].x = lY[o + k0];
            ya[blk][kc].y = lY[o + k1];
        }
    }

    v8f res[5];
#pragma unroll
    for (int f = 0; f < 5; ++f) {
        // ---- horizontal blur: Out(16x16) = F(16x28-band) * Wh(28x16) ----
#pragma unroll
        for (int blk = 0; blk < 2; ++blk) {
            v8f acc = {};
#pragma unroll
            for (int kc = 0; kc < KCH; ++kc) {
                float x0 = xa[blk][kc].x, y0 = ya[blk][kc].x;
                float x1 = xa[blk][kc].y, y1 = ya[blk][kc].y;
                v2f a;
                a.x = (f == 0) ? x0 : (f == 1) ? y0 : (f == 2) ? x0 * x0
                      : (f == 3) ? y0 * y0 : x0 * y0;
                a.y = (f == 0) ? x1 : (f == 1) ? y1 : (f == 2) ? x1 * x1
                      : (f == 3) ? y1 * y1 : x1 * y1;
                acc = __builtin_amdgcn_wmma_f32_16x16x4_f32(
                    false, a, false, wb[kc], (short)0, acc, false, false);
            }
            const int rb = blk * 10;    // rows 10..15 written twice, same value
#pragma unroll
            for (int v = 0; v < 8; ++v)
                lH[(rb + v + 8 * half) * SH + cb + lm] = acc[v];
        }
        // ---- vertical blur: Out(16x16) = Wv(16x28-band) * H(28x16) ----
        // Per-wave private staging strip; DS ops are in-order within a wave,
        // so no barrier is needed between the stores above and these loads.
        // K-pad rows (k>25) clamp to row 25 (their vertical weight is 0).
        v8f acc = {};
#pragma unroll
        for (int kc = 0; kc < KCH; ++kc) {
            int k  = kc * 4 + 2 * half;
            int r0 = k     > 25 ? 25 : k;
            int r1 = k + 1 > 25 ? 25 : k + 1;
            v2f bf;
            bf.x = lH[r0 * SH + cb + lm];
            bf.y = lH[r1 * SH + cb + lm];
            acc = __builtin_amdgcn_wmma_f32_16x16x4_f32(
                false, wb[kc], false, bf, (short)0, acc, false, false);
        }
        res[f] = acc;
    }

    // ---- SSIM map + masked per-lane sum (C/D layout: row v+8*half, col lm) ----
    const float C1v = 1.0e-4f, C2v = 9.0e-4f, EPSv = 1.0e-8f;
    const int gcol = col0 + cb + lm;
    float sum = 0.0f;
#pragma unroll
    for (int v = 0; v < 8; ++v) {
        int grow = row0 + v + 8 * half;
        float mx = res[0][v], my = res[1][v];
        float sx  = res[2][v] - mx * mx;
        float sy  = res[3][v] - my * my;
        float sxy = res[4][v] - mx * my;
        float cs = (2.0f * sxy + C2v) / (sx + sy + C2v + EPSv);
        cs = fmaxf(cs, 0.0f);
        float lum = (2.0f * mx * my + C1v) / (mx * mx + my * my + C1v + EPSv);
        float val = lum * cs;
        if (grow < HO && gcol < WO) sum += val;
    }

    // wave32 butterfly reduce, then one partial per workgroup (deterministic).
#pragma unroll
    for (int off = 16; off > 0; off >>= 1) sum += __shfl_xor(sum, off, 32);
    if (lane == 0) wavesum[wid] = sum;
    __syncthreads();
    if (threadIdx.x == 0) {
        float t = 0.0f;
#pragma unroll
        for (int i = 0; i < NWAVE; ++i) t += wavesum[i];
        partial[(size_t)img * (TILES_Y * TILES_X) + blockIdx.y * TILES_X + blockIdx.x] = t;
    }
}

// Deterministic per-batch reduction of tile partials (fixed summation order).
__global__ __launch_bounds__(256)
void ssim_reduce_kernel(const float* __restrict__ partial, float* __restrict__ out,
                        int per_batch, float inv_count) {
    __shared__ float buf[256];
    const float* p = partial + (size_t)blockIdx.x * per_batch;
    float s = 0.0f;
    for (int i = threadIdx.x; i < per_batch; i += 256) s += p[i];
    buf[threadIdx.x] = s;
    __syncthreads();
    for (int off = 128; off > 0; off >>= 1) {
        if (threadIdx.x < off) buf[threadIdx.x] += buf[threadIdx.x + off];
        __syncthreads();
    }
    if (threadIdx.x == 0) out[blockIdx.x] = buf[0] * inv_count;
}

extern "C" void kernel_launch(void* const* d_in, const int* in_sizes, int n_in,
                              void* d_out, int out_size, void* d_ws, size_t ws_size,
                              hipStream_t stream) {
    const float* X   = (const float*)d_in[0];
    const float* Y   = (const float*)d_in[1];
    const float* win = (const float*)d_in[2];
    float* partial   = (float*)d_ws;           // 48*48*6 floats = 55 KB
    float* out       = (float*)d_out;          // [16]

    dim3 grid(TILES_X, TILES_Y, NIMG);
    ssim_tile_kernel<<<grid, 256, 0, stream>>>(X, Y, win, partial);

    const int per_batch = CHAN * TILES_Y * TILES_X;          // 864
    const float inv_count = 1.0f / ((float)CHAN * HO * WO);  // mean over (C,H',W')
    ssim_reduce_kernel<<<BATCH, 256, 0, stream>>>(partial, out, per_batch, inv_count);
}